// ImportanceScoreHead_71021579207123
// MI455X (gfx1250) — compile-verified
//
#include <hip/hip_runtime.h>
#include <math.h>

// Problem constants from the reference setup (fixed shapes).
#define B_ 64
#define T_ 4096
#define D_ 128
#define K_ 512

typedef __attribute__((ext_vector_type(2))) float v2f;
typedef __attribute__((ext_vector_type(8))) float v8f;

__device__ __forceinline__ unsigned monoKey(float f) {
  // Order-preserving map: larger float -> larger unsigned.
  unsigned u = __float_as_uint(f);
  return (u & 0x80000000u) ? ~u : (u | 0x80000000u);
}

// ---------------------------------------------------------------------------
// Kernel A: scores[b,t] = dot(emb[b,t,:], W) + b0, masked -> -inf.
// Each wave computes 16 token scores via 32x V_WMMA_F32_16X16X4_F32.
// A = 16 tokens x 4 dims, B = 4 dims x 16 (all columns = same W chunk),
// so D[:,0] accumulates the 16 dot products. Lane layout per ISA:
//   lanes 0-15 hold K={0,1}, lanes 16-31 hold K={2,3} for both A and B.
// ---------------------------------------------------------------------------
__global__ __launch_bounds__(256) void score_kernel(
    const float* __restrict__ emb, const unsigned char* __restrict__ mask,
    const float* __restrict__ W, const float* __restrict__ bptr,
    float* __restrict__ scores) {
  __shared__ float sW[D_];
  const int tid = threadIdx.x;
  if (tid < D_) sW[tid] = W[tid];
  __syncthreads();

  const int lane = tid & 31;
  const int half = lane >> 4;      // 0: K={0,1}, 1: K={2,3}
  const int m    = lane & 15;      // token row within the 16x16 tile
  const long long waveId    = ((long long)blockIdx.x * blockDim.x + tid) >> 5;
  const long long tokenBase = waveId * 16;
  const float* row = emb + (tokenBase + m) * D_;
  const int koff = 2 * half;

  v8f acc = {};
#pragma unroll
  for (int kb = 0; kb < D_; kb += 4) {
    const float2 av = *(const float2*)(row + kb + koff);
    const float2 wv = *(const float2*)(sW + kb + koff);
    v2f a, bv;
    a.x = av.x;  a.y = av.y;
    bv.x = wv.x; bv.y = wv.y;
    acc = __builtin_amdgcn_wmma_f32_16x16x4_f32(
        /*neg_a=*/false, a, /*neg_b=*/false, bv,
        /*c_mod=*/(short)0, acc, /*reuse_a=*/false, /*reuse_b=*/false);
  }

  // D layout: VGPR r = M=r (lanes 0-15) / M=r+8 (lanes 16-31).
  // Lane 0 holds scores of tokens 0..7, lane 16 holds tokens 8..15.
  if (m == 0) {
    const float badd = bptr[0];
    const long long tb = tokenBase + (half ? 8 : 0);
#pragma unroll
    for (int i = 0; i < 8; ++i) {
      float v = acc[i] + badd;
      if (!mask[tb + i]) v = -__builtin_inff();
      scores[tb + i] = v;
    }
  }
}

// ---------------------------------------------------------------------------
// Kernel B: per batch row, exact top-K threshold via 4-pass radix select
// over monotonic keys in LDS, then ordered compaction of selected indices
// (ascending token order == sorted top-k indices; ties -> lowest index,
// matching jax.lax.top_k).
// One block of 256 threads per batch row; 16 contiguous tokens per thread.
// ---------------------------------------------------------------------------
__global__ __launch_bounds__(256) void topk_kernel(
    const float* __restrict__ scores, int* __restrict__ idxOut) {
  __shared__ unsigned keys[T_];
  __shared__ unsigned bins[256];
  __shared__ unsigned scanArr[256];
  __shared__ unsigned sh_prefix, sh_remaining;

  const int b   = blockIdx.x;
  const int tid = threadIdx.x;
  const float* srow = scores + (long long)b * T_;

  for (int i = tid; i < T_; i += 256) keys[i] = monoKey(srow[i]);
  __syncthreads();

  unsigned prefix = 0, prefmask = 0, remaining = K_;
  const int base = tid * 16;

  for (int pass = 3; pass >= 0; --pass) {
    bins[tid] = 0;
    __syncthreads();
#pragma unroll
    for (int i = 0; i < 16; ++i) {
      unsigned k = keys[base + i];
      if ((k & prefmask) == prefix)
        atomicAdd(&bins[(k >> (pass * 8)) & 255u], 1u);
    }
    __syncthreads();

    // Suffix sums: inclusive scan over reversed bins.
    scanArr[tid] = bins[255 - tid];
    __syncthreads();
    for (int off = 1; off < 256; off <<= 1) {
      unsigned v = (tid >= off) ? scanArr[tid - off] : 0u;
      __syncthreads();
      scanArr[tid] += v;
      __syncthreads();
    }
    const unsigned d      = 255u - (unsigned)tid;
    const unsigned suffix = scanArr[tid];      // sum of bins[j] for j >= d
    const unsigned bd     = bins[d];
    const unsigned gtc    = suffix - bd;       // strictly-greater count at this digit
    if (gtc < remaining && remaining <= suffix) {
      sh_prefix    = prefix | (d << (pass * 8));
      sh_remaining = remaining - gtc;
    }
    __syncthreads();
    prefix    = sh_prefix;
    remaining = sh_remaining;
    prefmask |= 0xFFu << (pass * 8);
    __syncthreads();
  }

  const unsigned thr    = prefix;     // key of K-th largest
  const unsigned needEq = remaining;  // how many ties to keep (lowest indices)

  // Ordered compaction: block scan of per-thread (gt, eq) counts.
  unsigned gcnt = 0, ecnt = 0;
#pragma unroll
  for (int i = 0; i < 16; ++i) {
    unsigned k = keys[base + i];
    gcnt += (k > thr);
    ecnt += (k == thr);
  }
  const unsigned packed = (gcnt << 16) | ecnt;
  scanArr[tid] = packed;
  __syncthreads();
  for (int off = 1; off < 256; off <<= 1) {
    unsigned v = (tid >= off) ? scanArr[tid - off] : 0u;
    __syncthreads();
    scanArr[tid] += v;
    __syncthreads();
  }
  const unsigned excl = scanArr[tid] - packed;
  unsigned gtB = excl >> 16;       // #gt tokens before my range
  unsigned eqB = excl & 0xFFFFu;   // #eq tokens before my range

  int* orow = idxOut + b * K_;
  for (int i = 0; i < 16; ++i) {
    const unsigned k = keys[base + i];
    if (k > thr) {
      const unsigned pos = gtB + (eqB < needEq ? eqB : needEq);
      orow[pos] = base + i;
      ++gtB;
    } else if (k == thr) {
      if (eqB < needEq) orow[gtB + eqB] = base + i;
      ++eqB;
    }
  }
}

// ---------------------------------------------------------------------------
// Kernel C: gather selected rows. One wave per (b, j): 32 lanes x float4.
// Applies the straight-through factor (1 - s) + s (s = sigmoid(score)) for
// bit-level fidelity with the reference (factor is 1.0 within 1 ulp).
// ---------------------------------------------------------------------------
__global__ __launch_bounds__(256) void gather_kernel(
    const float* __restrict__ emb, const unsigned char* __restrict__ mask,
    const float* __restrict__ scores, const int* __restrict__ idx,
    float* __restrict__ outSel, float* __restrict__ outMask) {
  const int tid  = threadIdx.x;
  const int lane = tid & 31;
  const long long w = ((long long)blockIdx.x * blockDim.x + tid) >> 5; // row id in [0, B*K)
  const int b = (int)(w >> 9);  // w / K_
  const int t = idx[w];

  const long long src_row = (long long)b * T_ + t;
  const float sc = scores[src_row];
  const float s  = 1.0f / (1.0f + __expf(-sc));  // sigmoid(-inf) == 0
  const float factor = (1.0f - s) + s;           // STE forward value

  float4 v = *(const float4*)(emb + src_row * D_ + lane * 4);
  v.x *= factor; v.y *= factor; v.z *= factor; v.w *= factor;
  *(float4*)(outSel + w * D_ + lane * 4) = v;

  if (lane == 0) outMask[w] = mask[src_row] ? 1.0f : 0.0f;
}

// ---------------------------------------------------------------------------
extern "C" void kernel_launch(void* const* d_in, const int* in_sizes, int n_in,
                              void* d_out, int out_size, void* d_ws, size_t ws_size,
                              hipStream_t stream) {
  (void)in_sizes; (void)n_in; (void)out_size; (void)ws_size;
  const float*         emb  = (const float*)d_in[0];
  const unsigned char* mask = (const unsigned char*)d_in[1];  // jax bool = 1 byte
  // d_in[2] is k (== 512 < T, so the k>=T early-return branch never triggers)
  const float*         W    = (const float*)d_in[3];
  const float*         bp   = (const float*)d_in[4];

  float* outSel  = (float*)d_out;                       // (B, K, D)
  float* outMask = outSel + (size_t)B_ * K_ * D_;       // (B, K)

  float* scores = (float*)d_ws;                                          // B*T floats
  int*   idx    = (int*)((char*)d_ws + (size_t)B_ * T_ * sizeof(float)); // B*K ints

  // A: scoring. waves = B*T/16; 8 waves/block.
  const int scoreThreads = (B_ * T_ / 16) * 32;
  score_kernel<<<scoreThreads / 256, 256, 0, stream>>>(emb, mask, W, bp, scores);

  // B: per-row exact top-K with ordered index output.
  topk_kernel<<<B_, 256, 0, stream>>>(scores, idx);

  // C: gather. waves = B*K; 8 waves/block.
  const int gatherThreads = (B_ * K_) * 32;
  gather_kernel<<<gatherThreads / 256, 256, 0, stream>>>(emb, mask, scores, idx,
                                                         outSel, outMask);
}